// NeighborlistBruteNsq_76115410419920
// MI455X (gfx1250) — compile-verified
//
#include <hip/hip_runtime.h>
#include <cstdint>
#include <math.h>

// ---------------------------------------------------------------------------
// Brute-force N^2 neighborlist with PBC (minimum image), N=4096, cutoff=0.5.
// Store-bandwidth-bound (~235 MB out). One wave32 == one 16x16 (i,j) tile:
//   - stage the 16-row i-tile and 16-row j-tile of positions into LDS via
//     global_load_async_to_lds_b128 (ASYNCcnt), s_wait_asynccnt 0
//   - one V_WMMA_F32_16X16X4_F32 computes the unwrapped d^2 tile:
//       A row m = (x_m, y_m, z_m, 1), B col n = (-2x_n, -2y_n, -2z_n, |p_n|^2),
//       C[m][n] = |p_m|^2   =>  D = |p_m|^2 + |p_n|^2 - 2 p_m.p_n
//   - per-pair VALU applies the min-image wrap (needed anyway for the r_ij
//     output) and corrects: d^2 = D + sum_c (w_c^2 - r_c^2)
//   - outputs streamed with non-temporal stores.
// ---------------------------------------------------------------------------

typedef float v2f __attribute__((ext_vector_type(2)));
typedef float v8f __attribute__((ext_vector_type(8)));

#define WAVES_PER_BLOCK 8
#define CUTOFF 0.5f

#if __has_builtin(__builtin_amdgcn_wmma_f32_16x16x4_f32)
#define HAVE_WMMA4 1
#else
#define HAVE_WMMA4 0
#if defined(__AMDGCN__)
#warning "wmma_f32_16x16x4_f32 builtin not found on device pass; VALU fallback"
#endif
#endif

// tiles in row r of the upper-triangular tile enumeration start at TBASE(r)
#define TBASE(r) ((r) * Nt - ((r) * ((r)-1)) / 2)

__global__ __launch_bounds__(256) void nlist_wmma_kernel(
    const float* __restrict__ pos,   // [n,3]
    const float* __restrict__ box,   // [3,3] diagonal used
    float* __restrict__ out,         // 7P words
    int n, int Nt, int T)
{
    __shared__ float smem[WAVES_PER_BLOCK * 96];  // per wave: 48 f (i-tile) + 48 f (j-tile)

    const int lane = threadIdx.x & 31;
    const int wv   = threadIdx.x >> 5;
    const int t    = blockIdx.x * WAVES_PER_BLOCK + wv;
    if (t >= T) return;  // wave-uniform exit

    // ---- invert flat tile id t -> (ti, tj) with tj >= ti ----
    int ti = (int)((2.0 * Nt + 1.0 -
                    sqrt((2.0 * Nt + 1.0) * (2.0 * Nt + 1.0) - 8.0 * (double)t)) * 0.5);
    if (ti < 0) ti = 0;
    if (ti >= Nt) ti = Nt - 1;
    while (ti > 0 && TBASE(ti) > t) --ti;
    while (TBASE(ti + 1) <= t) ++ti;
    const int tj = ti + (t - TBASE(ti));

    // ---- async-stage the two 16-row position tiles into this wave's LDS ----
    float* ldsW = &smem[wv * 96];
    // low 32 bits of a flat LDS pointer are the LDS byte address (aperture in high bits)
    const unsigned ldsA = (unsigned)(uintptr_t)ldsW;
    if (lane < 12) {  // 48 dwords = 12 x b128, i-tile
        unsigned la = ldsA + (unsigned)lane * 16u;
        const float* g = pos + (size_t)ti * 48 + (size_t)lane * 4;
        asm volatile("global_load_async_to_lds_b128 %0, %1, off"
                     :: "v"(la), "v"(g) : "memory");
    }
    if (lane >= 16 && lane < 28) {  // j-tile in parallel
        unsigned la = ldsA + 192u + (unsigned)(lane - 16) * 16u;
        const float* g = pos + (size_t)tj * 48 + (size_t)(lane - 16) * 4;
        asm volatile("global_load_async_to_lds_b128 %0, %1, off"
                     :: "v"(la), "v"(g) : "memory");
    }
    asm volatile("s_wait_asynccnt 0" ::: "memory");  // own-wave LDS region: no barrier needed

    const int  jl = lane & 15;
    const bool hi = lane >= 16;

    // ---- WMMA operands (32-bit 16x4 A layout: v0 = K0|K2, v1 = K1|K3 by lane half) ----
    const float pix = ldsW[jl * 3 + 0], piy = ldsW[jl * 3 + 1], piz = ldsW[jl * 3 + 2];
    const float pjx = ldsW[48 + jl * 3 + 0], pjy = ldsW[48 + jl * 3 + 1], pjz = ldsW[48 + jl * 3 + 2];
    const float nrmj = pjx * pjx + pjy * pjy + pjz * pjz;

    v2f A, B;
    A.x = hi ? piz : pix;                 // K2 : K0
    A.y = hi ? 1.0f : piy;                // K3 : K1
    B.x = -2.0f * (hi ? pjz : pjx);       // K2 : K0
    B.y = hi ? nrmj : (-2.0f * pjy);      // K3 : K1

    v8f C;
#pragma unroll
    for (int k = 0; k < 8; ++k) {         // C[m][n] = |p_m|^2 ; VGPR k holds M=k / M=8+k
        const int row = (hi ? 8 : 0) + k;
        const float qx = ldsW[row * 3 + 0], qy = ldsW[row * 3 + 1], qz = ldsW[row * 3 + 2];
        C[k] = qx * qx + qy * qy + qz * qz;
    }

#if HAVE_WMMA4
    // D = A(16x4) * B(4x16) + C  ->  unwrapped squared distances for the tile
    v8f D = __builtin_amdgcn_wmma_f32_16x16x4_f32(
        /*neg_a=*/false, A, /*neg_b=*/false, B,
        /*c_mod=*/(short)0, C, /*reuse_a=*/false, /*reuse_b=*/false);
#else
    v8f D = C;  // unused in fallback math below
#endif

    // ---- per-pair wrap + outputs ----
    const float Lx = box[0], Ly = box[4], Lz = box[8];
    const float hx = 0.5f * Lx, hy = 0.5f * Ly, hz = 0.5f * Lz;
    const float iLx = 1.0f / Lx, iLy = 1.0f / Ly, iLz = 1.0f / Lz;

    const long long P = (long long)n * (n - 1) / 2;
    int*   outI = (int*)out;          // pair_indices row 0
    int*   outJ = outI + P;           // pair_indices row 1
    float* outD = out + 2 * P;        // d_ij  [P,1]
    float* outR = out + 3 * P;        // r_ij  [P,3]
    int*   outM = (int*)out + 6 * P;  // in_cutoff [P] (0/1 words)

#pragma unroll
    for (int k = 0; k < 8; ++k) {
        const int il = (hi ? 8 : 0) + k;      // C/D layout: lanes<16 -> M=k, lanes>=16 -> M=8+k
        const int i  = ti * 16 + il;
        const int j  = tj * 16 + jl;
        if (j <= i || i >= n || j >= n) continue;

        const float qx = ldsW[il * 3 + 0], qy = ldsW[il * 3 + 1], qz = ldsW[il * 3 + 2];
        const float rx = qx - pjx, ry = qy - pjy, rz = qz - pjz;
        // minimum image: w = remainder(r + L/2, L) - L/2 = r - L*floor((r + L/2)/L)
        const float wx = rx - Lx * floorf((rx + hx) * iLx);
        const float wy = ry - Ly * floorf((ry + hy) * iLy);
        const float wz = rz - Lz * floorf((rz + hz) * iLz);

#if HAVE_WMMA4
        const float corr = (wx * wx - rx * rx) + (wy * wy - ry * ry) + (wz * wz - rz * rz);
        float d2 = D[k] + corr;
#else
        float d2 = wx * wx + wy * wy + wz * wz;
#endif
        d2 = fmaxf(d2, 0.0f);
        const float dist = sqrtf(d2);
        const bool  in   = dist <= CUTOFF;

        const long long p = (long long)i * (n - 1) - (long long)i * (i - 1) / 2 + (j - i - 1);

        __builtin_nontemporal_store(i, outI + p);
        __builtin_nontemporal_store(j, outJ + p);
        __builtin_nontemporal_store(in ? dist : 0.0f, outD + p);
        __builtin_nontemporal_store(in ? wx : 0.0f, outR + 3 * p + 0);
        __builtin_nontemporal_store(in ? wy : 0.0f, outR + 3 * p + 1);
        __builtin_nontemporal_store(in ? wz : 0.0f, outR + 3 * p + 2);
        __builtin_nontemporal_store(in ? 1 : 0, outM + p);
    }
}

extern "C" void kernel_launch(void* const* d_in, const int* in_sizes, int n_in,
                              void* d_out, int out_size, void* d_ws, size_t ws_size,
                              hipStream_t stream) {
    (void)out_size; (void)d_ws; (void)ws_size;
    const float* pos = (const float*)d_in[0];  // [n,3] f32
    const float* box = (const float*)d_in[1];  // [3,3] f32
    const int n = (n_in >= 3) ? in_sizes[2] : in_sizes[0] / 3;  // 4096

    const int Nt = (n + 15) / 16;                       // 256 tiles per axis
    const long long T = (long long)Nt * (Nt + 1) / 2;   // 32896 upper-tri tiles
    const int blocks = (int)((T + WAVES_PER_BLOCK - 1) / WAVES_PER_BLOCK);

    nlist_wmma_kernel<<<blocks, 32 * WAVES_PER_BLOCK, 0, stream>>>(
        pos, box, (float*)d_out, n, Nt, (int)T);
}